// RoPECausalAttention_11476152615280
// MI455X (gfx1250) — compile-verified
//
#include <hip/hip_runtime.h>
#include <math.h>

typedef float v2f __attribute__((ext_vector_type(2)));
typedef float v8f __attribute__((ext_vector_type(8)));

#define HDIM 256
#define SEQ  2048

// ---------------------------------------------------------------------------
// Extract cos/sin from the block-diagonal rotary tensor r (2048,256,256):
//   cos(s,i) = r[s, 2i,   2i]
//   sin(s,i) = r[s, 2i+1, 2i]
// Avoids streaming the 512 MB dense tensor in the hot path (2 MB table).
// ---------------------------------------------------------------------------
__global__ __launch_bounds__(128) void cs_kernel(const float* __restrict__ r,
                                                 float2* __restrict__ cs)
{
    int s = blockIdx.x;
    int i = threadIdx.x;                 // 0..127
    const float* rb = r + (size_t)s * HDIM * HDIM;
    float c  = rb[(size_t)(2 * i) * HDIM + 2 * i];
    float sn = rb[(size_t)(2 * i + 1) * HDIM + 2 * i];
    cs[s * (HDIM / 2) + i] = make_float2(c, sn);
}

// ---------------------------------------------------------------------------
// Projection GEMM:  out[row, n] = sum_h x[row, h] * w[n, h]   (rows = B*S)
// Two 16x16 C tiles per wave (one at a time) using V_WMMA_F32_16X16X4_F32 (exact fp32).
// RoPE fused into the epilogue for q (which=0) and k (which=1):
// column pairs (2i,2i+1) live in adjacent lanes -> one shfl_xor(1).
// ---------------------------------------------------------------------------
__global__ __launch_bounds__(256) void proj_kernel(const float* __restrict__ x,
    const float* __restrict__ wq, const float* __restrict__ wk,
    const float* __restrict__ wv, const float2* __restrict__ cs,
    float* __restrict__ qo, float* __restrict__ ko, float* __restrict__ vo)
{
    int rowTile = blockIdx.x;            // 16 rows each over B*S
    int which   = blockIdx.y;            // 0=q, 1=k, 2=v
    const float* w   = (which == 0) ? wq : ((which == 1) ? wk : wv);
    float*       out = (which == 0) ? qo : ((which == 1) ? ko : vo);

    int wave = threadIdx.x >> 5;         // 0..7
    int lane = threadIdx.x & 31;
    int lq   = lane & 15;
    int half = lane >> 4;
    int row0 = rowTile * 16;

    // A fragment: element (M = lq, K = h0 + 2*half + {0,1})
    const float* aRow = x + (size_t)(row0 + lq) * HDIM + 2 * half;

    for (int t = 0; t < 2; ++t) {
        int n0 = (wave + t * 8) * 16;
        // B fragment: element (K = h0 + 2*half + {0,1}, N = lq) = w[n0+lq][h]
        const float* bRow = w + (size_t)(n0 + lq) * HDIM + 2 * half;

        v8f acc = (v8f){0.f, 0.f, 0.f, 0.f, 0.f, 0.f, 0.f, 0.f};
#pragma unroll 8
        for (int h0 = 0; h0 < HDIM; h0 += 4) {
            v2f a = *(const v2f*)(aRow + h0);
            v2f b = *(const v2f*)(bRow + h0);
            acc = __builtin_amdgcn_wmma_f32_16x16x4_f32(
                false, a, false, b, (short)0, acc, false, false);
        }

        int n = n0 + lq;                 // output column (C layout: N = lane&15)
        if (which < 2) {
            int  i    = n >> 1;
            bool even = (n & 1) == 0;
#pragma unroll
            for (int vi = 0; vi < 8; ++vi) {
                int rowAbs = row0 + vi + 8 * half;       // = b*SEQ + s
                int s      = rowAbs & (SEQ - 1);
                float2 csv = cs[s * (HDIM / 2) + i];
                float own     = acc[vi];
                float partner = __shfl_xor(own, 1, 32);
                // even col 2i:   c*q[2i]   + s*q[2i+1]
                // odd  col 2i+1: c*q[2i+1] - s*q[2i]
                float res = even ? (csv.x * own + csv.y * partner)
                                 : (csv.x * own - csv.y * partner);
                out[(size_t)rowAbs * HDIM + n] = res;
            }
        } else {
#pragma unroll
            for (int vi = 0; vi < 8; ++vi) {
                int rowAbs = row0 + vi + 8 * half;
                out[(size_t)rowAbs * HDIM + n] = acc[vi];
            }
        }
    }
}

// ---------------------------------------------------------------------------
// Flash attention: one wave32 per (batch, 16-row q tile).
// S = Qr·Krᵀ / 16 via 64 f32 WMMAs; online softmax per C-layout row
// (row == one accumulator VGPR across a 16-lane half); P -> LDS to convert
// C layout to A layout; O += P·V via 64 f32 WMMAs across 16 column tiles.
// ---------------------------------------------------------------------------
__global__ __launch_bounds__(32) void attn_kernel(const float* __restrict__ qr,
    const float* __restrict__ kr, const float* __restrict__ vv,
    float* __restrict__ out)
{
    __shared__ float ldsQ[16 * HDIM];    // Q tile, row-major (16 KB)
    __shared__ float ldsP[16 * 16];      // P tile for layout conversion

    int qt   = blockIdx.x;               // 0..127
    int b    = blockIdx.y;               // 0..7
    int lane = threadIdx.x;
    int lq   = lane & 15;
    int half = lane >> 4;
    int q0   = qt * 16;

    const float* Qbase = qr + ((size_t)b * SEQ + q0) * HDIM;
    for (int idx = lane * 4; idx < 16 * HDIM; idx += 32 * 4)
        *(float4*)&ldsQ[idx] = *(const float4*)&Qbase[idx];
    __syncthreads();

    v8f O[16];
#pragma unroll
    for (int t = 0; t < 16; ++t) O[t] = (v8f){0.f,0.f,0.f,0.f,0.f,0.f,0.f,0.f};
    float mrow[8], lrow[8];
#pragma unroll
    for (int vi = 0; vi < 8; ++vi) { mrow[vi] = -__builtin_inff(); lrow[vi] = 0.f; }

    const float* Aq = &ldsQ[lq * HDIM + 2 * half];   // A frag base for Q

    for (int j = 0; j <= qt; ++j) {
        int k0 = j * 16;

        // ---- S tile = Q (16x256) · K (16x256)^T -------------------------
        const float* Kb = kr + ((size_t)b * SEQ + k0 + lq) * HDIM + 2 * half;
        v8f sacc = (v8f){0.f,0.f,0.f,0.f,0.f,0.f,0.f,0.f};
#pragma unroll 8
        for (int h0 = 0; h0 < HDIM; h0 += 4) {
            v2f a  = *(const v2f*)(Aq + h0);
            v2f bb = *(const v2f*)(Kb + h0);
            sacc = __builtin_amdgcn_wmma_f32_16x16x4_f32(
                false, a, false, bb, (short)0, sacc, false, false);
        }

        // ---- causal mask + online softmax (per row = per C vgpr/half) ---
        float alpha[8];
#pragma unroll
        for (int vi = 0; vi < 8; ++vi) {
            int m  = vi + 8 * half;
            int gq = q0 + m;
            int gk = k0 + lq;
            float sv = sacc[vi] * 0.0625f;              // * H^-0.5 = 1/16
            if (gk > gq) sv = -__builtin_inff();
            float rm = sv;                               // 16-lane row max
            rm = fmaxf(rm, __shfl_xor(rm, 1, 32));
            rm = fmaxf(rm, __shfl_xor(rm, 2, 32));
            rm = fmaxf(rm, __shfl_xor(rm, 4, 32));
            rm = fmaxf(rm, __shfl_xor(rm, 8, 32));
            float mnew = fmaxf(mrow[vi], rm);            // finite: diag unmasked
            float p  = expf(sv - mnew);
            float rs = p;                                // 16-lane row sum
            rs += __shfl_xor(rs, 1, 32);
            rs += __shfl_xor(rs, 2, 32);
            rs += __shfl_xor(rs, 4, 32);
            rs += __shfl_xor(rs, 8, 32);
            float a_ = expf(mrow[vi] - mnew);
            lrow[vi]  = lrow[vi] * a_ + rs;
            mrow[vi]  = mnew;
            alpha[vi] = a_;
            ldsP[m * 16 + lq] = p;                       // C layout -> LDS
        }
        __syncthreads();

        // ---- O = O*alpha + P (16x16) · V (16x256) -----------------------
        const float* Vb = vv + ((size_t)b * SEQ + k0) * HDIM + lq;
#pragma unroll
        for (int t = 0; t < 16; ++t) {
            v8f o = O[t];
#pragma unroll
            for (int vi = 0; vi < 8; ++vi) o[vi] *= alpha[vi];
            const float* Vt = Vb + t * 16;
#pragma unroll
            for (int kk = 0; kk < 16; kk += 4) {
                v2f a = *(const v2f*)&ldsP[lq * 16 + kk + 2 * half];  // A layout
                v2f bb;
                bb.x = Vt[(size_t)(kk + 2 * half + 0) * HDIM];
                bb.y = Vt[(size_t)(kk + 2 * half + 1) * HDIM];
                o = __builtin_amdgcn_wmma_f32_16x16x4_f32(
                    false, a, false, bb, (short)0, o, false, false);
            }
            O[t] = o;
        }
        __syncthreads();
    }

    // ---- epilogue: divide by row sum, store ----------------------------
    float* Ob = out + ((size_t)b * SEQ + q0) * HDIM;
#pragma unroll
    for (int vi = 0; vi < 8; ++vi) {
        int   m   = vi + 8 * half;
        float inv = 1.0f / lrow[vi];
#pragma unroll
        for (int t = 0; t < 16; ++t)
            Ob[(size_t)m * HDIM + t * 16 + lq] = O[t][vi] * inv;
    }
}

// ---------------------------------------------------------------------------
extern "C" void kernel_launch(void* const* d_in, const int* in_sizes, int n_in,
                              void* d_out, int out_size, void* d_ws, size_t ws_size,
                              hipStream_t stream)
{
    const float* x  = (const float*)d_in[0];
    const float* wq = (const float*)d_in[1];
    const float* wk = (const float*)d_in[2];
    const float* wv = (const float*)d_in[3];
    const float* r  = (const float*)d_in[4];
    float* out = (float*)d_out;

    const size_t nQKV = (size_t)8 * SEQ * HDIM;      // 4,194,304 floats each
    float*  q  = (float*)d_ws;
    float*  k  = q + nQKV;
    float*  v  = k + nQKV;
    float2* cs = (float2*)(v + nQKV);                // 2048*128 float2 (2 MB)

    cs_kernel<<<dim3(SEQ), dim3(HDIM / 2), 0, stream>>>(r, cs);
    proj_kernel<<<dim3((8 * SEQ) / 16, 3), dim3(256), 0, stream>>>(
        x, wq, wk, wv, cs, q, k, v);
    attn_kernel<<<dim3(SEQ / 16, 8), dim3(32), 0, stream>>>(q, k, v, out);
}